// GATNet_5420248728145
// MI455X (gfx1250) — compile-verified
//
#include <hip/hip_runtime.h>
#include <hip/hip_bf16.h>
#include <math.h>

// ---------------- problem constants (match reference) ----------------
#define N_NODES   262144
#define N_EDGES   4194304
#define E_TOT     (N_EDGES + N_NODES)   // edges + self loops
#define F_IN      4
#define HID       8
#define HEADS     2
#define N_CLS     3
#define NEG_SLOPE 0.2f
#define GAT_EPS   1e-16f

typedef __attribute__((ext_vector_type(2))) float v2f;
typedef __attribute__((ext_vector_type(8))) float v8f;

// ------------- ordered-uint encoding for float atomicMax -------------
__device__ __forceinline__ unsigned fenc(float f) {
    unsigned u = __float_as_uint(f);
    return (u & 0x80000000u) ? ~u : (u | 0x80000000u);
}
__device__ __forceinline__ float fdec(unsigned u) {
    return __uint_as_float((u & 0x80000000u) ? (u ^ 0x80000000u) : ~u);
}
__device__ __forceinline__ float leaky(float v) {
    return v > 0.f ? v : NEG_SLOPE * v;
}

// =====================================================================
// Layer-1 projection: h1[N,16] = x[N,4] @ W1[4,16]
// One V_WMMA_F32_16X16X4_F32 per 16 nodes (exact shape match).
// A layout (16x4 f32): lane m = lane&15; half = lane>>4;
//   a.x = A[m][2*half], a.y = A[m][2*half+1]
// C/D layout: vgpr r -> row (r + 8*half), col = lane&15
// =====================================================================
__global__ void proj1_wmma(const float* __restrict__ x,
                           const float* __restrict__ W1,
                           float* __restrict__ h1) {
    int wave = (blockIdx.x * blockDim.x + threadIdx.x) >> 5;
    int lane = threadIdx.x & 31;
    int col  = lane & 15;            // also row index m for A
    int half = lane >> 4;
    int base = wave * 16;            // 16 nodes per wave

    // A: 16 node rows x 4 features (contiguous float2)
    const float* xr = x + (size_t)(base + col) * F_IN + 2 * half;
    v2f a; a.x = xr[0]; a.y = xr[1];
    // B: W1 rows K = 2*half, 2*half+1 ; cols = lane&15
    v2f b; b.x = W1[(2 * half) * 16 + col];
           b.y = W1[(2 * half + 1) * 16 + col];
    v8f c = {};
    c = __builtin_amdgcn_wmma_f32_16x16x4_f32(false, a, false, b,
                                              (short)0, c, false, false);
    #pragma unroll
    for (int r = 0; r < 8; ++r)
        h1[(size_t)(base + r + 8 * half) * 16 + col] = c[r];
}

// =====================================================================
// Layer-2 projection with fused ReLU: g2[N,3] = relu(out1[N,16]) @ W2[16,3]
// K=16 done as four K=4 WMMAs accumulating in C. W2 zero-padded to 16 cols.
// =====================================================================
__global__ void proj2_wmma(const float* __restrict__ out1,
                           const float* __restrict__ W2,
                           float* __restrict__ g2) {
    int wave = (blockIdx.x * blockDim.x + threadIdx.x) >> 5;
    int lane = threadIdx.x & 31;
    int col  = lane & 15;
    int half = lane >> 4;
    int base = wave * 16;

    v8f acc = {};
    #pragma unroll
    for (int k = 0; k < 4; ++k) {
        int k0 = 4 * k + 2 * half;
        const float* hr = out1 + (size_t)(base + col) * 16 + k0;
        v2f a;
        a.x = fmaxf(hr[0], 0.f);     // ReLU fused into A load
        a.y = fmaxf(hr[1], 0.f);
        v2f b;
        b.x = (col < N_CLS) ? W2[(k0)     * N_CLS + col] : 0.f;
        b.y = (col < N_CLS) ? W2[(k0 + 1) * N_CLS + col] : 0.f;
        acc = __builtin_amdgcn_wmma_f32_16x16x4_f32(false, a, false, b,
                                                    (short)0, acc, false, false);
    }
    if (col < N_CLS) {
        #pragma unroll
        for (int r = 0; r < 8; ++r)
            g2[(size_t)(base + r + 8 * half) * N_CLS + col] = acc[r];
    }
}

// ------------- per-node attention-logit prep + accumulator init ------
__global__ void node_init1(const float* __restrict__ h1,
                           const float* __restrict__ a_src1,
                           const float* __restrict__ a_dst1,
                           float* __restrict__ alS, float* __restrict__ alD,
                           unsigned* __restrict__ m1, float* __restrict__ den1,
                           float* __restrict__ out1) {
    int n = blockIdx.x * blockDim.x + threadIdx.x;
    if (n >= N_NODES) return;
    #pragma unroll
    for (int hh = 0; hh < HEADS; ++hh) {
        float s = 0.f, d = 0.f;
        #pragma unroll
        for (int f = 0; f < HID; ++f) {
            float v = h1[(size_t)n * 16 + hh * HID + f];
            s += v * a_src1[hh * HID + f];
            d += v * a_dst1[hh * HID + f];
        }
        alS[n * HEADS + hh]  = s;
        alD[n * HEADS + hh]  = d;
        m1[n * HEADS + hh]   = fenc(-INFINITY);
        den1[n * HEADS + hh] = 0.f;
    }
    #pragma unroll
    for (int f = 0; f < 16; ++f) out1[(size_t)n * 16 + f] = 0.f;
}

__global__ void node_init2(const float* __restrict__ g2,
                           const float* __restrict__ a_src2,
                           const float* __restrict__ a_dst2,
                           float* __restrict__ alS, float* __restrict__ alD,
                           unsigned* __restrict__ m2, float* __restrict__ den2,
                           float* __restrict__ out2) {
    int n = blockIdx.x * blockDim.x + threadIdx.x;
    if (n >= N_NODES) return;
    float s = 0.f, d = 0.f;
    #pragma unroll
    for (int c = 0; c < N_CLS; ++c) {
        float v = g2[(size_t)n * N_CLS + c];
        s += v * a_src2[c];
        d += v * a_dst2[c];
        out2[(size_t)n * N_CLS + c] = 0.f;
    }
    alS[n] = s; alD[n] = d;
    m2[n] = fenc(-INFINITY); den2[n] = 0.f;
}

// ---------------------- edge passes (templated on heads) -------------
__device__ __forceinline__ void edge_sd(int e, const int* __restrict__ src,
                                        const int* __restrict__ dst,
                                        int& s, int& d) {
    if (e < N_EDGES) { s = src[e]; d = dst[e]; }
    else             { s = d = e - N_EDGES; }   // self loop
}

template <int H>
__global__ void edge_max_k(const int* __restrict__ src, const int* __restrict__ dst,
                           const float* __restrict__ alS, const float* __restrict__ alD,
                           unsigned* __restrict__ m) {
    int e = blockIdx.x * blockDim.x + threadIdx.x;
    if (e >= E_TOT) return;
    int s, d; edge_sd(e, src, dst, s, d);
    #pragma unroll
    for (int hh = 0; hh < H; ++hh) {
        float v = leaky(alS[s * H + hh] + alD[d * H + hh]);
        atomicMax(&m[d * H + hh], fenc(v));
    }
}

template <int H>
__global__ void edge_sum_k(const int* __restrict__ src, const int* __restrict__ dst,
                           const float* __restrict__ alS, const float* __restrict__ alD,
                           const unsigned* __restrict__ m, float* __restrict__ den) {
    int e = blockIdx.x * blockDim.x + threadIdx.x;
    if (e >= E_TOT) return;
    int s, d; edge_sd(e, src, dst, s, d);
    #pragma unroll
    for (int hh = 0; hh < H; ++hh) {
        float v  = leaky(alS[s * H + hh] + alD[d * H + hh]);
        float ex = __expf(v - fdec(m[d * H + hh]));
        atomicAdd(&den[d * H + hh], ex);     // native global_atomic_add_f32
    }
}

// F = features per head; feat layout [n][H][F]; out layout [n][H][F]
template <int H, int F>
__global__ void edge_acc_k(const int* __restrict__ src, const int* __restrict__ dst,
                           const float* __restrict__ alS, const float* __restrict__ alD,
                           const unsigned* __restrict__ m, const float* __restrict__ den,
                           const float* __restrict__ feat, float* __restrict__ out) {
    int e = blockIdx.x * blockDim.x + threadIdx.x;
    if (e >= E_TOT) return;
    int s, d; edge_sd(e, src, dst, s, d);
    float alpha[H];
    #pragma unroll
    for (int hh = 0; hh < H; ++hh) {
        float v  = leaky(alS[s * H + hh] + alD[d * H + hh]);
        float ex = __expf(v - fdec(m[d * H + hh]));
        alpha[hh] = ex / (den[d * H + hh] + GAT_EPS);
    }
    #pragma unroll
    for (int hh = 0; hh < H; ++hh)
        #pragma unroll
        for (int f = 0; f < F; ++f)
            atomicAdd(&out[((size_t)d * H + hh) * F + f],
                      feat[((size_t)s * H + hh) * F + f] * alpha[hh]);
}

// ---------------------------- final softmax --------------------------
__global__ void softmax3(const float* __restrict__ out2, float* __restrict__ y) {
    int n = blockIdx.x * blockDim.x + threadIdx.x;
    if (n >= N_NODES) return;
    float v0 = out2[(size_t)n * 3 + 0];
    float v1 = out2[(size_t)n * 3 + 1];
    float v2 = out2[(size_t)n * 3 + 2];
    float mx = fmaxf(v0, fmaxf(v1, v2));
    float e0 = __expf(v0 - mx), e1 = __expf(v1 - mx), e2 = __expf(v2 - mx);
    float inv = 1.f / (e0 + e1 + e2);
    y[(size_t)n * 3 + 0] = e0 * inv;
    y[(size_t)n * 3 + 1] = e1 * inv;
    y[(size_t)n * 3 + 2] = e2 * inv;
}

// =====================================================================
extern "C" void kernel_launch(void* const* d_in, const int* in_sizes, int n_in,
                              void* d_out, int out_size, void* d_ws, size_t ws_size,
                              hipStream_t stream) {
    const float* x      = (const float*)d_in[0];
    const int*   ei     = (const int*)  d_in[1];   // edge_index flattened [2,E] (int32)
    const float* W1     = (const float*)d_in[2];
    const float* a_src1 = (const float*)d_in[3];
    const float* a_dst1 = (const float*)d_in[4];
    const float* W2     = (const float*)d_in[5];
    const float* a_src2 = (const float*)d_in[6];
    const float* a_dst2 = (const float*)d_in[7];
    float* y = (float*)d_out;

    const int* src = ei;
    const int* dst = ei + N_EDGES;

    // ---- workspace carve-out (~52 MB, fits in L2) ----
    float* ws   = (float*)d_ws;
    float* h1   = ws;                 ws += (size_t)N_NODES * 16;
    float* alS1 = ws;                 ws += (size_t)N_NODES * HEADS;
    float* alD1 = ws;                 ws += (size_t)N_NODES * HEADS;
    unsigned* m1 = (unsigned*)ws;     ws += (size_t)N_NODES * HEADS;
    float* den1 = ws;                 ws += (size_t)N_NODES * HEADS;
    float* out1 = ws;                 ws += (size_t)N_NODES * 16;
    float* g2   = ws;                 ws += (size_t)N_NODES * N_CLS;
    float* alS2 = ws;                 ws += (size_t)N_NODES;
    float* alD2 = ws;                 ws += (size_t)N_NODES;
    unsigned* m2 = (unsigned*)ws;     ws += (size_t)N_NODES;
    float* den2 = ws;                 ws += (size_t)N_NODES;
    float* out2 = ws;                 ws += (size_t)N_NODES * N_CLS;

    const int TB = 256;                       // 8 waves / block (wave32)
    const int gridProj = N_NODES / (16 * 8);  // 16 nodes per wave
    const int gridNode = (N_NODES + TB - 1) / TB;
    const int gridEdge = (E_TOT   + TB - 1) / TB;

    // ---------------- layer 1 ----------------
    proj1_wmma<<<gridProj, TB, 0, stream>>>(x, W1, h1);
    node_init1<<<gridNode, TB, 0, stream>>>(h1, a_src1, a_dst1,
                                            alS1, alD1, m1, den1, out1);
    edge_max_k<HEADS><<<gridEdge, TB, 0, stream>>>(src, dst, alS1, alD1, m1);
    edge_sum_k<HEADS><<<gridEdge, TB, 0, stream>>>(src, dst, alS1, alD1, m1, den1);
    edge_acc_k<HEADS, HID><<<gridEdge, TB, 0, stream>>>(src, dst, alS1, alD1,
                                                        m1, den1, h1, out1);

    // ---------------- layer 2 ----------------
    proj2_wmma<<<gridProj, TB, 0, stream>>>(out1, W2, g2);   // ReLU fused
    node_init2<<<gridNode, TB, 0, stream>>>(g2, a_src2, a_dst2,
                                            alS2, alD2, m2, den2, out2);
    edge_max_k<1><<<gridEdge, TB, 0, stream>>>(src, dst, alS2, alD2, m2);
    edge_sum_k<1><<<gridEdge, TB, 0, stream>>>(src, dst, alS2, alD2, m2, den2);
    edge_acc_k<1, N_CLS><<<gridEdge, TB, 0, stream>>>(src, dst, alS2, alD2,
                                                      m2, den2, g2, out2);

    // ---------------- output softmax ----------------
    softmax3<<<gridNode, TB, 0, stream>>>(out2, y);
}